// HyperGraphCustomBipartiteDisenGATVAEV3CTRObjSameIdxHyperGraph_1975684956803
// MI455X (gfx1250) — compile-verified
//
#include <hip/hip_runtime.h>

#define N_NODES 1000000
#define N_EDGES 8000000
#define EMB 64
#define CDIM 32
#define NEG_SLOPE 0.01f
#define EPS_ 1e-10f

typedef __attribute__((ext_vector_type(2))) float v2f;
typedef __attribute__((ext_vector_type(8))) float v8f;

__device__ __forceinline__ float leaky(float x) { return x > 0.f ? x : NEG_SLOPE * x; }

// Monotonic float -> uint mapping so unsigned atomicMax == float max.
__device__ __forceinline__ unsigned fflip(float f) {
    unsigned u = __float_as_uint(f);
    return (u & 0x80000000u) ? ~u : (u | 0x80000000u);
}
__device__ __forceinline__ float funflip(unsigned k) {
    return __uint_as_float((k & 0x80000000u) ? (k ^ 0x80000000u) : ~k);
}

__global__ void zero_kernel(unsigned* __restrict__ p, long long n) {
    long long i = (long long)blockIdx.x * blockDim.x + threadIdx.x;
    long long stride = (long long)gridDim.x * blockDim.x;
    for (; i < n; i += stride) p[i] = 0u;
}

// ---------------------------------------------------------------------------
// Projection: h[c] = leaky_relu(ego @ W[c] + b[c])  via V_WMMA_F32_16X16X4_F32
// One wave computes a 16-node x 32-col tile for BOTH channels.
// A 16x4 layout:  lanes 0-15 -> M=lane, VGPR{0,1}=K{0,1}; lanes 16-31 -> K{2,3}
// B 4x16 layout:  lanes 0-15 -> N=lane, VGPR{0,1}=K{0,1}; lanes 16-31 -> K{2,3}
// D 16x16 layout: VGPR r: lanes 0-15 -> M=r, lanes 16-31 -> M=r+8; N=lane%16
// ---------------------------------------------------------------------------
#define A_STRIDE 68   // 64 + 4 pad floats: bank-conflict-free, float4-aligned

__global__ __launch_bounds__(256)
void proj_kernel(const float* __restrict__ ego, const float* __restrict__ W,
                 const float* __restrict__ bias, float* __restrict__ h)
{
    __shared__ float sW[2 * EMB * CDIM];        // 16 KB
    __shared__ float sB[2 * CDIM];
    __shared__ float sA[8 * 16 * A_STRIDE];     // 8 waves * 16 rows * 68 floats

    const int tid = threadIdx.x;
    for (int i = tid; i < 2 * EMB * CDIM; i += 256) sW[i] = W[i];
    for (int i = tid; i < 2 * CDIM; i += 256)       sB[i] = bias[i];

    const int wave  = tid >> 5;
    const int lane  = tid & 31;
    const int mrow  = lane & 15;
    const int khalf = lane >> 4;            // 0: lanes 0-15 (K 0,1), 1: lanes 16-31 (K 2,3)

    const int nTiles = N_NODES / 16;        // 62500 exactly
    const int tile   = blockIdx.x * 8 + wave;
    float* sAw = &sA[wave * 16 * A_STRIDE];

    if (tile < nTiles) {
        // Stage 16x64 A tile (coalesced 128b loads), padded rows in LDS.
        const float4* src4 = reinterpret_cast<const float4*>(ego + (size_t)tile * 16 * EMB);
        float4* dst4 = reinterpret_cast<float4*>(sAw);
        #pragma unroll
        for (int i = 0; i < 8; ++i) {
            int idx4 = i * 32 + lane;       // 0..255 flat float4 index of tile
            int arow = idx4 >> 4;           // 16 float4 per row
            int ac4  = idx4 & 15;
            dst4[arow * (A_STRIDE / 4) + ac4] = src4[idx4];
        }
    }
    __syncthreads();
    if (tile >= nTiles) return;

    v8f acc[2][2];
    #pragma unroll
    for (int c = 0; c < 2; ++c)
        #pragma unroll
        for (int nh = 0; nh < 2; ++nh)
            acc[c][nh] = (v8f){};

    #pragma unroll
    for (int kk = 0; kk < 16; ++kk) {
        const int k0 = kk * 4 + 2 * khalf;
        v2f a;
        a[0] = sAw[mrow * A_STRIDE + k0];
        a[1] = sAw[mrow * A_STRIDE + k0 + 1];
        #pragma unroll
        for (int c = 0; c < 2; ++c) {
            const float* wc = &sW[c * EMB * CDIM];   // [64][32] row-major: W[k][n]
            #pragma unroll
            for (int nh = 0; nh < 2; ++nh) {
                const int ncol = nh * 16 + mrow;
                v2f bf;
                bf[0] = wc[k0 * CDIM + ncol];
                bf[1] = wc[(k0 + 1) * CDIM + ncol];
                acc[c][nh] = __builtin_amdgcn_wmma_f32_16x16x4_f32(
                    false, a, false, bf, (short)0, acc[c][nh], false, false);
            }
        }
    }

    // Write back: bias + leaky_relu, h[c] is [N][32] row-major.
    const int nodeBase = tile * 16 + 8 * khalf;
    #pragma unroll
    for (int c = 0; c < 2; ++c) {
        float* hc = h + (size_t)c * N_NODES * CDIM;
        #pragma unroll
        for (int nh = 0; nh < 2; ++nh) {
            const int ncol = nh * 16 + mrow;
            const float bv = sB[c * CDIM + ncol];
            #pragma unroll
            for (int r = 0; r < 8; ++r) {
                hc[(size_t)(nodeBase + r) * CDIM + ncol] = leaky(acc[c][nh][r] + bv);
            }
        }
    }
}

// ---------------------------------------------------------------------------
// Edge logits: 8 lanes per edge; each lane loads float4 of the 32-float rows
// (one 128B cacheline per row), dot + 3-step xor-shuffle reduction.
// ---------------------------------------------------------------------------
__global__ __launch_bounds__(256)
void edge_logit_kernel(const float* __restrict__ h, const int* __restrict__ row,
                       const int* __restrict__ col, float* __restrict__ logits,
                       unsigned* __restrict__ rowmax)
{
    const long long gtid = (long long)blockIdx.x * blockDim.x + threadIdx.x;
    const long long e = gtid >> 3;
    const int sub = (int)(gtid & 7);
    if (e >= N_EDGES) return;
    const int r  = row[e];
    const int cx = col[e];

    const float4* h0 = reinterpret_cast<const float4*>(h);
    const float4* h1 = reinterpret_cast<const float4*>(h + (size_t)N_NODES * CDIM);
    float4 a0 = h0[(size_t)r  * 8 + sub];
    float4 b0 = h0[(size_t)cx * 8 + sub];
    float4 a1 = h1[(size_t)r  * 8 + sub];
    float4 b1 = h1[(size_t)cx * 8 + sub];

    float p0 = a0.x * b0.x + a0.y * b0.y + a0.z * b0.z + a0.w * b0.w;
    float p1 = a1.x * b1.x + a1.y * b1.y + a1.z * b1.z + a1.w * b1.w;
    #pragma unroll
    for (int off = 4; off > 0; off >>= 1) {   // masks 1,2,4 stay inside the 8-lane group
        p0 += __shfl_xor(p0, off);
        p1 += __shfl_xor(p1, off);
    }
    if (sub == 0) {
        float l0 = leaky(p0);
        float l1 = leaky(p1);
        logits[e]           = l0;
        logits[N_EDGES + e] = l1;
        atomicMax(&rowmax[r],           fflip(l0));
        atomicMax(&rowmax[N_NODES + r], fflip(l1));
    }
}

// ---------------------------------------------------------------------------
// exp(logit - rowmax) and denominator accumulation (segment_sum via f32 atomics)
// ---------------------------------------------------------------------------
__global__ __launch_bounds__(256)
void edge_exp_kernel(const int* __restrict__ row, float* __restrict__ logits,
                     const unsigned* __restrict__ rowmax, float* __restrict__ denom)
{
    long long e = (long long)blockIdx.x * blockDim.x + threadIdx.x;
    if (e >= N_EDGES) return;
    int r = row[e];
    float m0 = funflip(rowmax[r]);
    float m1 = funflip(rowmax[N_NODES + r]);
    float e0 = expf(logits[e] - m0);
    float e1 = expf(logits[N_EDGES + e] - m1);
    logits[e]           = e0;   // reuse buffer as "ex"
    logits[N_EDGES + e] = e1;
    atomicAdd(&denom[r],           e0);
    atomicAdd(&denom[N_NODES + r], e1);
}

// ---------------------------------------------------------------------------
// Weighted scatter: out[r, c*32+j] += alpha_c * h_c[col, j]; 8 lanes per edge.
// ---------------------------------------------------------------------------
__global__ __launch_bounds__(256)
void edge_scatter_kernel(const float* __restrict__ h, const int* __restrict__ row,
                         const int* __restrict__ col, const float* __restrict__ ex,
                         const float* __restrict__ denom, float* __restrict__ out)
{
    const long long gtid = (long long)blockIdx.x * blockDim.x + threadIdx.x;
    const long long e = gtid >> 3;
    const int sub = (int)(gtid & 7);
    if (e >= N_EDGES) return;
    const int r  = row[e];
    const int cx = col[e];

    const float alpha0 = ex[e]           / (denom[r]           + EPS_);
    const float alpha1 = ex[N_EDGES + e] / (denom[N_NODES + r] + EPS_);

    const float4* h0 = reinterpret_cast<const float4*>(h);
    const float4* h1 = reinterpret_cast<const float4*>(h + (size_t)N_NODES * CDIM);
    float4 c0 = h0[(size_t)cx * 8 + sub];
    float4 c1 = h1[(size_t)cx * 8 + sub];

    float* o0 = out + (size_t)r * EMB + sub * 4;        // channel 0 -> cols [0,32)
    float* o1 = o0 + CDIM;                              // channel 1 -> cols [32,64)
    atomicAdd(o0 + 0, alpha0 * c0.x);
    atomicAdd(o0 + 1, alpha0 * c0.y);
    atomicAdd(o0 + 2, alpha0 * c0.z);
    atomicAdd(o0 + 3, alpha0 * c0.w);
    atomicAdd(o1 + 0, alpha1 * c1.x);
    atomicAdd(o1 + 1, alpha1 * c1.y);
    atomicAdd(o1 + 2, alpha1 * c1.z);
    atomicAdd(o1 + 3, alpha1 * c1.w);
}

extern "C" void kernel_launch(void* const* d_in, const int* in_sizes, int n_in,
                              void* d_out, int out_size, void* d_ws, size_t ws_size,
                              hipStream_t stream)
{
    const float* ego = (const float*)d_in[0];   // [N, 64]
    const float* W   = (const float*)d_in[1];   // [2, 64, 32]
    const float* b   = (const float*)d_in[2];   // [2, 1, 32]
    const int*   row = (const int*)d_in[3];     // [E]
    const int*   col = (const int*)d_in[4];     // [E]
    float* out = (float*)d_out;                 // [N, 64]

    char* ws = (char*)d_ws;
    float* h = (float*)ws;                                           // 2*N*32 f32 (256 MB)
    size_t off = (size_t)2 * N_NODES * CDIM * sizeof(float);
    float* logits = (float*)(ws + off);                              // 2*E f32 (64 MB)
    off += (size_t)2 * N_EDGES * sizeof(float);
    unsigned* rowmax = (unsigned*)(ws + off);                        // 2*N u32 (8 MB)
    off += (size_t)2 * N_NODES * sizeof(unsigned);
    float* denom = (float*)(ws + off);                               // 2*N f32 (8 MB)

    // Initialize accumulators (0u == 0.0f bits; rowmax flip-encoding identity is 0u).
    zero_kernel<<<2048, 256, 0, stream>>>((unsigned*)out, (long long)N_NODES * EMB);
    zero_kernel<<<512, 256, 0, stream>>>(rowmax, (long long)2 * N_NODES);
    zero_kernel<<<512, 256, 0, stream>>>((unsigned*)denom, (long long)2 * N_NODES);

    const int nTiles = N_NODES / 16;                     // 62500
    proj_kernel<<<(nTiles + 7) / 8, 256, 0, stream>>>(ego, W, b, h);

    const long long edgeLanes = (long long)N_EDGES * 8;
    edge_logit_kernel<<<(int)((edgeLanes + 255) / 256), 256, 0, stream>>>(h, row, col, logits, rowmax);
    edge_exp_kernel<<<(N_EDGES + 255) / 256, 256, 0, stream>>>(row, logits, rowmax, denom);
    edge_scatter_kernel<<<(int)((edgeLanes + 255) / 256), 256, 0, stream>>>(h, row, col, logits, denom, out);
}